// HeteroRGCN_14224931684972
// MI455X (gfx1250) — compile-verified
//
#include <hip/hip_runtime.h>

typedef float v2f __attribute__((ext_vector_type(2)));
typedef float v8f __attribute__((ext_vector_type(8)));

// ---------------------------------------------------------------------------
// Zero-fill (float4 per thread)
// ---------------------------------------------------------------------------
__global__ __launch_bounds__(256) void zero4_kernel(float4* __restrict__ p, int n4) {
    int i = blockIdx.x * blockDim.x + threadIdx.x;
    if (i < n4) p[i] = make_float4(0.f, 0.f, 0.f, 0.f);
}

// ---------------------------------------------------------------------------
// Per-etype in-degree count (float so we can invert in place)
// ---------------------------------------------------------------------------
__global__ __launch_bounds__(256) void deg_count_kernel(const int* __restrict__ dst,
                                                        float* __restrict__ deg, int E) {
    int i = blockIdx.x * blockDim.x + threadIdx.x;
    if (i < E) atomicAdd(&deg[dst[i]], 1.0f);
}

__global__ __launch_bounds__(256) void inv_deg_kernel(float* __restrict__ deg, int n) {
    int i = blockIdx.x * blockDim.x + threadIdx.x;
    if (i < n) deg[i] = 1.0f / fmaxf(deg[i], 1.0f);
}

__global__ __launch_bounds__(256) void leaky_relu_kernel(float* __restrict__ x, int n) {
    int i = blockIdx.x * blockDim.x + threadIdx.x;
    if (i < n) {
        float v = x[i];
        x[i] = v > 0.f ? v : 0.01f * v;
    }
}

// ---------------------------------------------------------------------------
// C[M,N] = A[M,128] @ W[128,N] + bias   via V_WMMA_F32_16X16X4_F32
// One wave computes a 16-row strip across all N (NT 16-col accumulators).
// Tile guard is wave-uniform -> EXEC all ones inside WMMA path.
// ---------------------------------------------------------------------------
template <int NT>
__global__ __launch_bounds__(256) void gemm_bias_wmma_kernel(
    const float* __restrict__ A, const float* __restrict__ W,
    const float* __restrict__ bias, float* __restrict__ C, int mtiles) {
    constexpr int N = NT * 16;
    constexpr int K = 128;

    int wid  = threadIdx.x >> 5;
    int tile = blockIdx.x * 8 + wid;
    if (tile >= mtiles) return;                 // wave-uniform exit

    int lane = threadIdx.x & 31;
    int half = lane >> 4;                       // 0: K pair {0,1}, 1: K pair {2,3}
    int lrow = lane & 15;

    // A fragment source: row = tile*16 + lrow, k offset 2*half
    const float* arow = A + (size_t)(tile * 16 + lrow) * K + 2 * half;

    v8f acc[NT];
#pragma unroll
    for (int t = 0; t < NT; ++t) {
        float bv = bias[t * 16 + lrow];         // lane's column is fixed per tile
        acc[t] = (v8f){bv, bv, bv, bv, bv, bv, bv, bv};
    }

#pragma unroll 2
    for (int k = 0; k < K; k += 4) {
        v2f a = *(const v2f*)(arow + k);        // A[row, k+2h], A[row, k+2h+1]
        const float* wr = W + (size_t)(k + 2 * half) * N + lrow;
#pragma unroll
        for (int t = 0; t < NT; ++t) {
            v2f b;
            b.x = wr[t * 16];                   // W[k+2h  , col]
            b.y = wr[N + t * 16];               // W[k+2h+1, col]
            acc[t] = __builtin_amdgcn_wmma_f32_16x16x4_f32(
                false, a, false, b, (short)0, acc[t], false, false);
        }
    }

    // C layout: lane column = lrow, rows = tile*16 + half*8 + v
    float* crow = C + (size_t)(tile * 16 + half * 8) * N + lrow;
#pragma unroll
    for (int t = 0; t < NT; ++t) {
#pragma unroll
        for (int v = 0; v < 8; ++v) crow[(size_t)v * N + t * 16] = acc[t][v];
    }
}

// ---------------------------------------------------------------------------
// Mean-aggregation scatter: one wave per edge.
// acc[dst] += Wh[src] * invdeg[dst]   (VPL floats per lane, d = 32*VPL)
// ---------------------------------------------------------------------------
template <int VPL>
__global__ __launch_bounds__(256) void scatter_mean_kernel(
    const float* __restrict__ Wh, const int* __restrict__ src,
    const int* __restrict__ dst, const float* __restrict__ invdeg,
    float* __restrict__ acc, int E) {
    int wid = blockIdx.x * 8 + (threadIdx.x >> 5);
    if (wid >= E) return;
    int lane = threadIdx.x & 31;

    int   s  = src[wid];
    int   dd = dst[wid];
    float w  = invdeg[dd];

    constexpr int D = VPL * 32;
    const float* p = Wh + (size_t)s * D + lane * VPL;
    float*       q = acc + (size_t)dd * D + lane * VPL;
#pragma unroll
    for (int j = 0; j < VPL; ++j) atomicAdd(&q[j], p[j] * w);
}

// ---------------------------------------------------------------------------
// Host driver
// ---------------------------------------------------------------------------
extern "C" void kernel_launch(void* const* d_in, const int* in_sizes, int n_in,
                              void* d_out, int out_size, void* d_ws, size_t ws_size,
                              hipStream_t stream) {
    const float* emb_user     = (const float*)d_in[0];
    const float* emb_item     = (const float*)d_in[1];
    const float* W1_rates     = (const float*)d_in[2];
    const float* b1_rates     = (const float*)d_in[3];
    const float* W1_rated_by  = (const float*)d_in[6];
    const float* b1_rated_by  = (const float*)d_in[7];
    const float* W2_rated_by  = (const float*)d_in[8];
    const float* b2_rated_by  = (const float*)d_in[9];
    const float* W1_follows   = (const float*)d_in[10];
    const float* b1_follows   = (const float*)d_in[11];
    const float* W2_follows   = (const float*)d_in[12];
    const float* b2_follows   = (const float*)d_in[13];
    const float* W1_similar   = (const float*)d_in[14];
    const float* b1_similar   = (const float*)d_in[15];
    const int* src_rates      = (const int*)d_in[18];
    const int* dst_rates      = (const int*)d_in[19];
    const int* src_rated_by   = (const int*)d_in[20];
    const int* dst_rated_by   = (const int*)d_in[21];
    const int* src_follows    = (const int*)d_in[22];
    const int* dst_follows    = (const int*)d_in[23];
    const int* src_similar    = (const int*)d_in[24];
    const int* dst_similar    = (const int*)d_in[25];

    const int NU = in_sizes[0] / 128;   // 100000
    const int NI = in_sizes[1] / 128;   // 100000
    const int E  = in_sizes[18];        // 500000

    // Workspace layout (floats)
    float* Wh   = (float*)d_ws;                         // max(NU,NI)*128
    size_t whN  = (size_t)(NU > NI ? NU : NI) * 128;
    float* accU = Wh + whN;                             // NU*128
    float* accI = accU + (size_t)NU * 128;              // NI*128
    float* degRates = accI + (size_t)NI * 128;          // NI
    float* degRb    = degRates + NI;                    // NU
    float* degF     = degRb + NU;                       // NU
    float* degS     = degF + NU;                        // NI

    const int accTot = NU * 128 + NI * 128;             // accU..accI contiguous
    const int degTot = NI + NU + NU + NI;

    // ---- zero accumulators / degrees / output ----
    zero4_kernel<<<(accTot / 4 + 255) / 256, 256, 0, stream>>>((float4*)accU, accTot / 4);
    zero4_kernel<<<(degTot / 4 + 255) / 256, 256, 0, stream>>>((float4*)degRates, degTot / 4);
    zero4_kernel<<<(out_size / 4 + 255) / 256, 256, 0, stream>>>((float4*)d_out, out_size / 4);

    // ---- degrees + inverse (shared by both layers) ----
    int degBlocks = (E + 255) / 256;
    deg_count_kernel<<<degBlocks, 256, 0, stream>>>(dst_rates,    degRates, E);
    deg_count_kernel<<<degBlocks, 256, 0, stream>>>(dst_rated_by, degRb,    E);
    deg_count_kernel<<<degBlocks, 256, 0, stream>>>(dst_follows,  degF,     E);
    deg_count_kernel<<<degBlocks, 256, 0, stream>>>(dst_similar,  degS,     E);
    inv_deg_kernel<<<(degTot + 255) / 256, 256, 0, stream>>>(degRates, degTot);

    const int mtU = NU / 16, mtI = NI / 16;
    const int gU = (mtU + 7) / 8, gI = (mtI + 7) / 8;
    const int sBlocks = (E + 7) / 8;                    // one wave per edge

    // ---- Layer 1 (d = 128), etype-sequential so Wh scratch is reused ----
    gemm_bias_wmma_kernel<8><<<gU, 256, 0, stream>>>(emb_user, W1_rates, b1_rates, Wh, mtU);
    scatter_mean_kernel<4><<<sBlocks, 256, 0, stream>>>(Wh, src_rates, dst_rates, degRates, accI, E);

    gemm_bias_wmma_kernel<8><<<gI, 256, 0, stream>>>(emb_item, W1_rated_by, b1_rated_by, Wh, mtI);
    scatter_mean_kernel<4><<<sBlocks, 256, 0, stream>>>(Wh, src_rated_by, dst_rated_by, degRb, accU, E);

    gemm_bias_wmma_kernel<8><<<gU, 256, 0, stream>>>(emb_user, W1_follows, b1_follows, Wh, mtU);
    scatter_mean_kernel<4><<<sBlocks, 256, 0, stream>>>(Wh, src_follows, dst_follows, degF, accU, E);

    gemm_bias_wmma_kernel<8><<<gI, 256, 0, stream>>>(emb_item, W1_similar, b1_similar, Wh, mtI);
    scatter_mean_kernel<4><<<sBlocks, 256, 0, stream>>>(Wh, src_similar, dst_similar, degS, accI, E);

    // ---- leaky_relu over both node types (contiguous) ----
    leaky_relu_kernel<<<(accTot + 255) / 256, 256, 0, stream>>>(accU, accTot);

    // ---- Layer 2 (d = 64): only the two user-destined relations matter ----
    gemm_bias_wmma_kernel<4><<<gI, 256, 0, stream>>>(accI, W2_rated_by, b2_rated_by, Wh, mtI);
    scatter_mean_kernel<2><<<sBlocks, 256, 0, stream>>>(Wh, src_rated_by, dst_rated_by, degRb, (float*)d_out, E);

    gemm_bias_wmma_kernel<4><<<gU, 256, 0, stream>>>(accU, W2_follows, b2_follows, Wh, mtU);
    scatter_mean_kernel<2><<<sBlocks, 256, 0, stream>>>(Wh, src_follows, dst_follows, degF, (float*)d_out, E);
}